// ContextPicoSNN_23038204575753
// MI455X (gfx1250) — compile-verified
//
#include <hip/hip_runtime.h>
#include <hip/hip_bf16.h>

typedef float v2f __attribute__((ext_vector_type(2)));
typedef float v8f __attribute__((ext_vector_type(8)));

#define EPSF      1e-5f
#define TAU_E     0.92f
#define TAU_C     0.9f
#define VTHF      1.0f
#define CRWF      0.35f

__device__ __forceinline__ float silu_f(float x) { return x / (1.f + expf(-x)); }

__device__ __forceinline__ v8f wmma_k4(v2f a, v2f b, v8f c) {
    // D = A(16x4 f32) * B(4x16 f32) + C(16x16 f32), full f32 precision.
    return __builtin_amdgcn_wmma_f32_16x16x4_f32(false, a, false, b, (short)0, c, false, false);
}

// ---------------------------------------------------------------------------
// Kernel A: multi-scale stem -> tokens [2560, 40, 24] (LayerNormed)
// One block per sequence n in [0, 2560). 256 threads.
// LDS overlay: [0,4512) = sx(3000) then feat2(24x188); [4512,13512) = fused(12x750)
// ---------------------------------------------------------------------------
__global__ __launch_bounds__(256)
void stem_kernel(const float* __restrict__ x,
                 const float* __restrict__ wb1, const float* __restrict__ wb2, const float* __restrict__ wb3,
                 const float* __restrict__ gb1, const float* __restrict__ bb1,
                 const float* __restrict__ gb2, const float* __restrict__ bb2,
                 const float* __restrict__ gb3, const float* __restrict__ bb3,
                 const float* __restrict__ wfuse, const float* __restrict__ gfuse, const float* __restrict__ bfuse,
                 const float* __restrict__ wdw1, const float* __restrict__ wpw1,
                 const float* __restrict__ g1, const float* __restrict__ b1,
                 const float* __restrict__ wdw2, const float* __restrict__ wpw2,
                 const float* __restrict__ g2, const float* __restrict__ b2,
                 const float* __restrict__ gtn, const float* __restrict__ btn,
                 float* __restrict__ tokens)
{
    __shared__ float lds[13512];
    float* sx    = lds;          // 3000 floats (stage 1 only)
    float* feat2 = lds;          // 24*188 = 4512 floats (stage 3, after sx dead)
    float* fused = lds + 4512;   // 12*750 = 9000 floats

    const int tid = threadIdx.x;
    const int n   = blockIdx.x;
    const float* xg = x + (size_t)n * 3000;

    __builtin_prefetch(xg + tid, 0, 0);   // global_prefetch_b8
    for (int i = tid; i < 3000; i += 256) sx[i] = xg[i];
    __syncthreads();

    const float BNS = rsqrtf(1.f + EPSF); // eval-mode BN: running_var=1

    // ---- stage 1: 3 conv branches (s=4) + BN + SiLU + 1x1 fuse + BN + SiLU ----
    for (int t = tid; t < 750; t += 256) {
        float br[12];
        const int base = 4 * t;
        for (int oc = 0; oc < 4; ++oc) {                 // k=31, pad=15
            float a = 0.f;
            for (int j = 0; j < 31; ++j) {
                int p = base - 15 + j;
                if (p >= 0 && p < 3000) a += sx[p] * wb1[oc * 31 + j];
            }
            br[oc] = silu_f(a * (gb1[oc] * BNS) + bb1[oc]);
        }
        for (int oc = 0; oc < 4; ++oc) {                 // k=63, pad=31
            float a = 0.f;
            for (int j = 0; j < 63; ++j) {
                int p = base - 31 + j;
                if (p >= 0 && p < 3000) a += sx[p] * wb2[oc * 63 + j];
            }
            br[4 + oc] = silu_f(a * (gb2[oc] * BNS) + bb2[oc]);
        }
        for (int oc = 0; oc < 4; ++oc) {                 // k=125, pad=62
            float a = 0.f;
            for (int j = 0; j < 125; ++j) {
                int p = base - 62 + j;
                if (p >= 0 && p < 3000) a += sx[p] * wb3[oc * 125 + j];
            }
            br[8 + oc] = silu_f(a * (gb3[oc] * BNS) + bb3[oc]);
        }
        for (int oc = 0; oc < 12; ++oc) {                // 1x1 fuse 12->12
            float a = 0.f;
            for (int ic = 0; ic < 12; ++ic) a += br[ic] * wfuse[oc * 12 + ic];
            fused[oc * 750 + t] = silu_f(a * (gfuse[oc] * BNS) + bfuse[oc]);
        }
    }
    __syncthreads();

    // ---- stage 2+3 fused: dw1(k9,s2)+pw1(12->24)+BN+SiLU, dw2(k7,s2)+pw2(24->24)+BN+SiLU
    for (int t = tid; t < 188; t += 256) {
        float acc2[24];
        for (int oc = 0; oc < 24; ++oc) acc2[oc] = 0.f;
        for (int j = 0; j < 7; ++j) {                    // dw2 taps, p1 in pw1-output space (375)
            int p1 = 2 * t - 3 + j;
            if (p1 < 0 || p1 >= 375) continue;
            float dwv[12];
            for (int c = 0; c < 12; ++c) {               // dw1 at position p1
                float a = 0.f;
                for (int u = 0; u < 9; ++u) {
                    int q = 2 * p1 - 4 + u;
                    if (q >= 0 && q < 750) a += fused[c * 750 + q] * wdw1[c * 9 + u];
                }
                dwv[c] = a;
            }
            for (int oc = 0; oc < 24; ++oc) {            // pw1 + BN + SiLU, feed dw2 tap
                float a = 0.f;
                for (int c = 0; c < 12; ++c) a += dwv[c] * wpw1[oc * 12 + c];
                float f1 = silu_f(a * (g1[oc] * BNS) + b1[oc]);
                acc2[oc] += f1 * wdw2[oc * 7 + j];
            }
        }
        for (int o2 = 0; o2 < 24; ++o2) {                // pw2 + BN + SiLU
            float a = 0.f;
            for (int oc = 0; oc < 24; ++oc) a += acc2[oc] * wpw2[o2 * 24 + oc];
            feat2[o2 * 188 + t] = silu_f(a * (g2[o2] * BNS) + b2[o2]);
        }
    }
    __syncthreads();

    // ---- stage 4: linspace select 40 steps + token LayerNorm(24) ----
    if (tid < 40) {
        const int c = (tid * 187) / 39;                  // floor(i*187/39) == np.linspace int
        float vals[24];
        float m = 0.f;
        for (int d = 0; d < 24; ++d) { vals[d] = feat2[d * 188 + c]; m += vals[d]; }
        m *= (1.f / 24.f);
        float var = 0.f;
        for (int d = 0; d < 24; ++d) { float dd = vals[d] - m; var += dd * dd; }
        var *= (1.f / 24.f);
        const float rs = rsqrtf(var + EPSF);
        float* tout = tokens + (size_t)n * 960 + tid * 24;
        for (int d = 0; d < 24; ++d) tout[d] = (vals[d] - m) * rs * gtn[d] + btn[d];
    }
}

// ---------------------------------------------------------------------------
// Kernel B: recurrent spiking epoch encoder with f32 WMMA (16x16x4).
// 80 blocks x 64 threads (2 waves); each wave owns 16 batch rows, N=96 hidden.
// ---------------------------------------------------------------------------
__global__ __launch_bounds__(64)
void lif_encoder_kernel(const float* __restrict__ tokens,
                        const float* __restrict__ Wein,   // [96,24]
                        const float* __restrict__ Werec,  // [96,96]
                        const float* __restrict__ Weproj, // [48,96]
                        const float* __restrict__ gen, const float* __restrict__ ben,
                        float* __restrict__ emb)          // [2560,48]
{
    __shared__ float WrT[96 * 96];     // B-operand for recurrence: WrT[k*96+n] = Werec[n][k]
    __shared__ float WeinT[24 * 96];   // B-operand for currents:  WeinT[d*96+h] = Wein[h][d]
    __shared__ float sspk[2 * 16 * 96];
    __shared__ float pool[2 * 16 * 48];

    const int tid  = threadIdx.x;
    const int wv   = tid >> 5;
    const int lane = tid & 31;
    const int half = lane >> 4;
    const int lm   = lane & 15;
    const int koff = half * 2;                 // K offset for lanes 16..31 (ISA 16x4 A layout)
    const int n0   = blockIdx.x * 32 + wv * 16;

    for (int i = tid; i < 96 * 96; i += 64) { int k = i / 96, h = i % 96; WrT[i]   = Werec[h * 96 + k]; }
    for (int i = tid; i < 24 * 96; i += 64) { int d = i / 96, h = i % 96; WeinT[i] = Wein[h * 24 + d]; }
    for (int i = tid; i < 2 * 16 * 96; i += 64) sspk[i] = 0.f;
    __syncthreads();

    float* sl = sspk + wv * 16 * 96;
    float* pl = pool + wv * 16 * 48;

    v8f vmem[6], sreg[6], ssum[6];
    #pragma unroll
    for (int t = 0; t < 6; ++t) {
        vmem[t] = (v8f){0.f,0.f,0.f,0.f,0.f,0.f,0.f,0.f};
        sreg[t] = vmem[t];
        ssum[t] = vmem[t];
    }

    const float* tokrow = tokens + (size_t)(n0 + lm) * 960;

    for (int s = 0; s < 40; ++s) {
        v2f atok[6];                                   // A fragments: tokens[16 rows x 24]
        #pragma unroll
        for (int kt = 0; kt < 6; ++kt) {
            atok[kt].x = tokrow[s * 24 + kt * 4 + koff];
            atok[kt].y = tokrow[s * 24 + kt * 4 + koff + 1];
        }
        v2f aspk[24];                                  // A fragments: prev spikes [16 x 96]
        #pragma unroll
        for (int kt = 0; kt < 24; ++kt) {
            aspk[kt].x = sl[lm * 96 + kt * 4 + koff];
            aspk[kt].y = sl[lm * 96 + kt * 4 + koff + 1];
        }
        #pragma unroll
        for (int nt = 0; nt < 6; ++nt) {
            const int nn = nt * 16 + lm;
            v8f acc = (v8f){0.f,0.f,0.f,0.f,0.f,0.f,0.f,0.f};
            #pragma unroll
            for (int kt = 0; kt < 6; ++kt) {           // currents: tokens @ Wein^T
                v2f b;
                b.x = WeinT[(kt * 4 + koff)     * 96 + nn];
                b.y = WeinT[(kt * 4 + koff + 1) * 96 + nn];
                acc = wmma_k4(atok[kt], b, acc);
            }
            #pragma unroll
            for (int kt = 0; kt < 24; ++kt) {          // recurrence: s @ Wr^T
                v2f b;
                b.x = WrT[(kt * 4 + koff)     * 96 + nn];
                b.y = WrT[(kt * 4 + koff + 1) * 96 + nn];
                acc = wmma_k4(aspk[kt], b, acc);
            }
            #pragma unroll
            for (int i = 0; i < 8; ++i) {              // LIF update on C/D layout
                float vo = vmem[nt][i];
                float so = sreg[nt][i];
                float vn = TAU_E * vo * (1.f - so) + acc[i];
                float sn = (vn - VTHF > 0.f) ? 1.f : 0.f;
                vmem[nt][i] = vn;
                sreg[nt][i] = sn;
                ssum[nt][i] += sn;
                sl[(i + 8 * half) * 96 + nn] = sn;     // publish for next step's A operand
            }
        }
    }

    // ---- pooled = (sum_s spikes) @ Weproj^T / 40, then LayerNorm(48) -> emb ----
    #pragma unroll
    for (int nt = 0; nt < 6; ++nt)
        #pragma unroll
        for (int i = 0; i < 8; ++i)
            sl[(i + 8 * half) * 96 + nt * 16 + lm] = ssum[nt][i];
    __syncthreads();

    v2f aps[24];
    #pragma unroll
    for (int kt = 0; kt < 24; ++kt) {
        aps[kt].x = sl[lm * 96 + kt * 4 + koff];
        aps[kt].y = sl[lm * 96 + kt * 4 + koff + 1];
    }
    #pragma unroll
    for (int et = 0; et < 3; ++et) {
        const int ee = et * 16 + lm;
        v8f acc = (v8f){0.f,0.f,0.f,0.f,0.f,0.f,0.f,0.f};
        #pragma unroll
        for (int kt = 0; kt < 24; ++kt) {
            v2f b;
            b.x = Weproj[ee * 96 + kt * 4 + koff];     // B[k][e] = Weproj[e][k], from global (L2)
            b.y = Weproj[ee * 96 + kt * 4 + koff + 1];
            acc = wmma_k4(aps[kt], b, acc);
        }
        #pragma unroll
        for (int i = 0; i < 8; ++i)
            pl[(i + 8 * half) * 48 + ee] = acc[i] * (1.f / 40.f);
    }
    __syncthreads();

    if (half == 0) {                                   // lanes 0..15: LN row lm
        const int n = n0 + lm;
        float m = 0.f;
        for (int e = 0; e < 48; ++e) m += pl[lm * 48 + e];
        m *= (1.f / 48.f);
        float var = 0.f;
        for (int e = 0; e < 48; ++e) { float d = pl[lm * 48 + e] - m; var += d * d; }
        var *= (1.f / 48.f);
        const float rs = rsqrtf(var + EPSF);
        for (int e = 0; e < 48; ++e)
            emb[(size_t)n * 48 + e] = (pl[lm * 48 + e] - m) * rs * gen[e] + ben[e];
    }
}

// ---------------------------------------------------------------------------
// Kernel C: bidirectional context SNN over L=5 + fuse + head (center epoch).
// One wave32 per batch row (lane = hidden unit). 128 blocks x 128 threads.
// ---------------------------------------------------------------------------
__global__ __launch_bounds__(128)
void context_head_kernel(const float* __restrict__ emb,   // [512,5,48]
                         const float* __restrict__ Wcfi, const float* __restrict__ Wcfr,
                         const float* __restrict__ Wcbi, const float* __restrict__ Wcbr,
                         const float* __restrict__ Wcproj,
                         const float* __restrict__ gcn, const float* __restrict__ bcn,
                         const float* __restrict__ Wfus,  // [48,112]
                         const float* __restrict__ gfn, const float* __restrict__ bfn,
                         const float* __restrict__ Wh, const float* __restrict__ bh,
                         float* __restrict__ out)         // [512,5]
{
    __shared__ float cEmb[4][240];
    __shared__ float cS[4][32];
    __shared__ float cCf[4][32];
    __shared__ float cCb[4][32];
    __shared__ float cFus[4][48];

    const int wv   = threadIdx.x >> 5;
    const int lane = threadIdx.x & 31;
    const int b    = blockIdx.x * 4 + wv;

    for (int i = lane; i < 240; i += 32) cEmb[wv][i] = emb[(size_t)b * 240 + i];
    __syncthreads();

    // ---- forward scan l = 0..2 (only sp_f[center] needed) ----
    float v = 0.f, sp = 0.f;
    for (int l = 0; l <= 2; ++l) {
        float cur = 0.f;
        for (int d = 0; d < 48; ++d) cur += cEmb[wv][l * 48 + d] * Wcfi[lane * 48 + d];
        cS[wv][lane] = sp;
        __syncthreads();
        float rec = 0.f;
        for (int h = 0; h < 32; ++h) rec += cS[wv][h] * Wcfr[lane * 32 + h];
        __syncthreads();
        v  = TAU_C * v * (1.f - sp) + cur + rec;
        sp = (v - VTHF > 0.f) ? 1.f : 0.f;
    }
    cS[wv][lane] = sp;
    __syncthreads();
    {
        float c = 0.f;
        for (int h = 0; h < 32; ++h) c += cS[wv][h] * Wcproj[lane * 32 + h];
        cCf[wv][lane] = c;
    }
    __syncthreads();
    {   // LayerNorm(32) of ctx_f
        float m = 0.f;
        for (int h = 0; h < 32; ++h) m += cCf[wv][h];
        m *= (1.f / 32.f);
        float var = 0.f;
        for (int h = 0; h < 32; ++h) { float d = cCf[wv][h] - m; var += d * d; }
        var *= (1.f / 32.f);
        const float nc = (cCf[wv][lane] - m) * rsqrtf(var + EPSF) * gcn[lane] + bcn[lane];
        __syncthreads();
        cCf[wv][lane] = nc;
    }
    __syncthreads();

    // ---- backward scan l = 4..2 ----
    v = 0.f; sp = 0.f;
    for (int idx = 0; idx <= 2; ++idx) {
        const int l = 4 - idx;
        float cur = 0.f;
        for (int d = 0; d < 48; ++d) cur += cEmb[wv][l * 48 + d] * Wcbi[lane * 48 + d];
        cS[wv][lane] = sp;
        __syncthreads();
        float rec = 0.f;
        for (int h = 0; h < 32; ++h) rec += cS[wv][h] * Wcbr[lane * 32 + h];
        __syncthreads();
        v  = TAU_C * v * (1.f - sp) + cur + rec;
        sp = (v - VTHF > 0.f) ? 1.f : 0.f;
    }
    cS[wv][lane] = sp;
    __syncthreads();
    {
        float c = 0.f;
        for (int h = 0; h < 32; ++h) c += cS[wv][h] * Wcproj[lane * 32 + h];
        cCb[wv][lane] = c;
    }
    __syncthreads();
    {   // LayerNorm(32) of ctx_b
        float m = 0.f;
        for (int h = 0; h < 32; ++h) m += cCb[wv][h];
        m *= (1.f / 32.f);
        float var = 0.f;
        for (int h = 0; h < 32; ++h) { float d = cCb[wv][h] - m; var += d * d; }
        var *= (1.f / 32.f);
        const float nc = (cCb[wv][lane] - m) * rsqrtf(var + EPSF) * gcn[lane] + bcn[lane];
        __syncthreads();
        cCb[wv][lane] = nc;
    }
    __syncthreads();

    // ---- fused = silu(LN48(fused_in @ Wfus^T)); fused_in = [emb_c, ctx_f, ctx_b] ----
    const int o0 = lane, o1 = lane + 32;
    float fr0 = 0.f, fr1 = 0.f;
    for (int d = 0; d < 48; ++d) fr0 += cEmb[wv][96 + d] * Wfus[o0 * 112 + d];
    for (int h = 0; h < 32; ++h) fr0 += cCf[wv][h] * Wfus[o0 * 112 + 48 + h];
    for (int h = 0; h < 32; ++h) fr0 += cCb[wv][h] * Wfus[o0 * 112 + 80 + h];
    if (lane < 16) {
        for (int d = 0; d < 48; ++d) fr1 += cEmb[wv][96 + d] * Wfus[o1 * 112 + d];
        for (int h = 0; h < 32; ++h) fr1 += cCf[wv][h] * Wfus[o1 * 112 + 48 + h];
        for (int h = 0; h < 32; ++h) fr1 += cCb[wv][h] * Wfus[o1 * 112 + 80 + h];
    }
    cFus[wv][o0] = fr0;
    if (lane < 16) cFus[wv][o1] = fr1;
    __syncthreads();

    float m = 0.f;
    for (int o = 0; o < 48; ++o) m += cFus[wv][o];
    m *= (1.f / 48.f);
    float var = 0.f;
    for (int o = 0; o < 48; ++o) { float d = cFus[wv][o] - m; var += d * d; }
    var *= (1.f / 48.f);
    const float rs = rsqrtf(var + EPSF);
    const float f0 = silu_f((fr0 - m) * rs * gfn[o0] + bfn[o0]);
    const float f1 = (lane < 16) ? silu_f((fr1 - m) * rs * gfn[o1] + bfn[o1]) : 0.f;
    __syncthreads();
    cFus[wv][o0] = f0;
    if (lane < 16) cFus[wv][o1] = f1;
    __syncthreads();

    if (lane < 5) {
        float a = bh[lane];
        for (int o = 0; o < 48; ++o)
            a += (cFus[wv][o] + CRWF * cEmb[wv][96 + o]) * Wh[lane * 48 + o];
        out[(size_t)b * 5 + lane] = a;
    }
}

// ---------------------------------------------------------------------------
extern "C" void kernel_launch(void* const* d_in, const int* in_sizes, int n_in,
                              void* d_out, int out_size, void* d_ws, size_t ws_size,
                              hipStream_t stream) {
    const float* x      = (const float*)d_in[0];
    const float* wb1    = (const float*)d_in[1];
    const float* wb2    = (const float*)d_in[2];
    const float* wb3    = (const float*)d_in[3];
    const float* gb1    = (const float*)d_in[4];
    const float* bb1    = (const float*)d_in[5];
    const float* gb2    = (const float*)d_in[6];
    const float* bb2    = (const float*)d_in[7];
    const float* gb3    = (const float*)d_in[8];
    const float* bb3    = (const float*)d_in[9];
    const float* wfuse  = (const float*)d_in[10];
    const float* gfuse  = (const float*)d_in[11];
    const float* bfuse  = (const float*)d_in[12];
    const float* wdw1   = (const float*)d_in[13];
    const float* wpw1   = (const float*)d_in[14];
    const float* g1     = (const float*)d_in[15];
    const float* b1     = (const float*)d_in[16];
    const float* wdw2   = (const float*)d_in[17];
    const float* wpw2   = (const float*)d_in[18];
    const float* g2     = (const float*)d_in[19];
    const float* b2     = (const float*)d_in[20];
    const float* gtn    = (const float*)d_in[21];
    const float* btn    = (const float*)d_in[22];
    const float* Wein   = (const float*)d_in[23];
    const float* Werec  = (const float*)d_in[24];
    const float* Weproj = (const float*)d_in[25];
    const float* gen    = (const float*)d_in[26];
    const float* ben    = (const float*)d_in[27];
    const float* Wcfi   = (const float*)d_in[28];
    const float* Wcfr   = (const float*)d_in[29];
    const float* Wcbi   = (const float*)d_in[30];
    const float* Wcbr   = (const float*)d_in[31];
    const float* Wcproj = (const float*)d_in[32];
    const float* gcn    = (const float*)d_in[33];
    const float* bcn    = (const float*)d_in[34];
    const float* Wfus   = (const float*)d_in[35];
    const float* gfn    = (const float*)d_in[36];
    const float* bfn    = (const float*)d_in[37];
    const float* Wh     = (const float*)d_in[38];
    const float* bhp    = (const float*)d_in[39];

    float* wsf    = (float*)d_ws;
    float* tokens = wsf;                       // 2560 * 40 * 24 floats
    float* embb   = wsf + (size_t)2560 * 960;  // 2560 * 48 floats

    stem_kernel<<<2560, 256, 0, stream>>>(x, wb1, wb2, wb3, gb1, bb1, gb2, bb2, gb3, bb3,
                                          wfuse, gfuse, bfuse, wdw1, wpw1, g1, b1,
                                          wdw2, wpw2, g2, b2, gtn, btn, tokens);

    lif_encoder_kernel<<<80, 64, 0, stream>>>(tokens, Wein, Werec, Weproj, gen, ben, embb);

    context_head_kernel<<<128, 128, 0, stream>>>(embb, Wcfi, Wcfr, Wcbi, Wcbr, Wcproj,
                                                 gcn, bcn, Wfus, gfn, bfn, Wh, bhp,
                                                 (float*)d_out);
}